// RNN_49211735277960
// MI455X (gfx1250) — compile-verified
//
#include <hip/hip_runtime.h>
#include <hip/hip_bf16.h>

// Problem constants (from reference)
constexpr int B_ = 128;   // batch
constexpr int S_ = 1024;  // seq
constexpr int D_ = 512;   // din
constexpr int H_ = 512;   // hid

typedef __attribute__((ext_vector_type(2))) float v2f;
typedef __attribute__((ext_vector_type(4))) float v4f;
typedef __attribute__((ext_vector_type(8))) float v8f;

// Swizzled LDS index for the h tile: conflict-free b64 A-operand loads without
// padding (4-dword blocks rotated per row; col%4 stays intact so the b64 at
// col = k + 2*halfsel (k % 4 == 0) never straddles a block).
__device__ __forceinline__ int hidx(int row, int col) {
    return row * H_ + (col & 3) + ((((col >> 2) + row) & 127) << 2);
}

// -----------------------------------------------------------------------------
// Phase 0: pack a 512x512 weight matrix into WMMA-B operand order.
// Record (k8, nt, lane) -> 4 dwords: { W[k+2h][n], W[k+1+2h][n],           (chunk k)
//                                      W[k+4+2h][n], W[k+5+2h][n] }        (chunk k+4)
// where k = 8*k8, h = lane>>4, n = nt*16 + (lane&15). One b128 load then feeds
// two V_WMMA_F32_16X16X4_F32 ops.
// -----------------------------------------------------------------------------
__global__ __launch_bounds__(128) void pack_w_kernel(
    const float* __restrict__ W, float* __restrict__ Wp)
{
    const int lane = threadIdx.x & 31;
    const int wave = threadIdx.x >> 5;
    const int task = blockIdx.x * 4 + wave;   // 0..2047 = (k8, nt)
    const int k8   = task >> 5;               // 0..63
    const int nt   = task & 31;               // 0..31
    const int l15  = lane & 15;
    const int h    = lane >> 4;
    const int col  = nt * 16 + l15;
    const int k    = k8 * 8;

    v4f d;
    d[0] = W[(size_t)(k     + 2 * h) * H_ + col];
    d[1] = W[(size_t)(k + 1 + 2 * h) * H_ + col];
    d[2] = W[(size_t)(k + 4 + 2 * h) * H_ + col];
    d[3] = W[(size_t)(k + 5 + 2 * h) * H_ + col];
    *(v4f*)(Wp + ((size_t)(k8 * 32 + nt) * 32 + lane) * 4) = d;
}

// -----------------------------------------------------------------------------
// Phase 1: XP = x@Wx + bias, emitted in packed C-tile layout.
// M-tile = 16 batch rows at fixed s (so scan tiles match), N-tile = 16 cols.
// Packed tile t=(s, bt, nt): 256 dwords at base t*256, lane stores its 8
// accumulator dwords contiguously at lane*8 (two b128 stores).
// Grid (s=1024, bt=8, ngrp=8), 4 waves/block, wave -> nt = ngrp*4 + wave.
// -----------------------------------------------------------------------------
__global__ __launch_bounds__(128) void xp_gemm_kernel(
    const float* __restrict__ x, const float* __restrict__ wxp,
    const float* __restrict__ bias, float* __restrict__ xp_p)
{
    const int lane    = threadIdx.x & 31;
    const int wave    = threadIdx.x >> 5;
    const int l15     = lane & 15;
    const int halfsel = lane >> 4;

    const int s  = blockIdx.x;
    const int bt = blockIdx.y;
    const int nt = blockIdx.z * 4 + wave;

    // A: x[b = bt*16 + l15, s, k + 2h .. +1]  (b64 per chunk)
    const float* aptr = x + ((size_t)(bt * 16 + l15) * S_ + s) * D_ + 2 * halfsel;
    const float* wptr = wxp + ((size_t)nt * 32 + lane) * 4;   // += 4096/k8

    const float bv = bias[nt * 16 + l15];
    v8f c = {bv, bv, bv, bv, bv, bv, bv, bv};

    #pragma unroll 4
    for (int k8 = 0; k8 < D_ / 8; ++k8) {
        const int k = k8 * 8;
        v2f a0 = *(const v2f*)(aptr + k);
        v2f a1 = *(const v2f*)(aptr + k + 4);
        v4f w  = *(const v4f*)(wptr + (size_t)k8 * 4096);
        c = __builtin_amdgcn_wmma_f32_16x16x4_f32(false, a0, false,
                __builtin_shufflevector(w, w, 0, 1), (short)0, c, false, false);
        c = __builtin_amdgcn_wmma_f32_16x16x4_f32(false, a1, false,
                __builtin_shufflevector(w, w, 2, 3), (short)0, c, false, false);
    }

    float* dst = xp_p + (((size_t)s * 8 + bt) * 32 + nt) * 256 + lane * 8;
    *(v4f*)(dst)     = __builtin_shufflevector(c, c, 0, 1, 2, 3);
    *(v4f*)(dst + 4) = __builtin_shufflevector(c, c, 4, 5, 6, 7);
}

// -----------------------------------------------------------------------------
// Phase 2: persistent scan. 8 blocks x 16 waves; block bt owns batch rows
// [bt*16, bt*16+16). h tile lives in swizzled LDS; Wh streams from L2 in packed
// b128 form; the next step's 32 KB XP slab is DMA'd global->LDS with
// global_load_async_to_lds_b128 (per-wave private 2 KB region, ASYNCcnt-paced)
// while the current step's WMMAs execute.
// -----------------------------------------------------------------------------
__global__ __launch_bounds__(512) void rnn_scan_kernel(
    const float* __restrict__ xp_p, const float* __restrict__ whp,
    float* __restrict__ out)
{
    __shared__ float hbuf[16 * H_];      // 32 KB, swizzled via hidx()
    __shared__ float xstage[32 * 256];   // 32 KB, one step's packed XP tiles

    const int lane    = threadIdx.x & 31;
    const int wave    = threadIdx.x >> 5;     // 0..15
    const int l15     = lane & 15;
    const int halfsel = lane >> 4;
    const int bt      = blockIdx.x;
    const int bbase   = bt * 16;
    const int nt0     = wave * 2;             // this wave's two N-tiles
    const int n0      = nt0 * 16;
    const int n1      = n0 + 16;

    for (int i = threadIdx.x; i < 16 * H_; i += blockDim.x) hbuf[i] = 0.0f;
    __syncthreads();

    const float* wb0 = whp + ((size_t)nt0 * 32 + lane) * 4;         // tile nt0
    const float* wb1 = whp + ((size_t)(nt0 + 1) * 32 + lane) * 4;   // tile nt0+1
    const float* stg = xstage + nt0 * 256 + lane * 8;               // stage read ptr

    for (int s = 0; s < S_; ++s) {
        // ---- accumulator init: c = XP tile (exact C layout, b128 moves) ----
        v8f c0, c1;
        if (s == 0) {
            const float* g = xp_p + ((size_t)bt * 32 + nt0) * 256 + lane * 8;
            v4f a = *(const v4f*)(g);       v4f b = *(const v4f*)(g + 4);
            v4f e = *(const v4f*)(g + 256); v4f f = *(const v4f*)(g + 260);
            c0 = __builtin_shufflevector(a, b, 0, 1, 2, 3, 4, 5, 6, 7);
            c1 = __builtin_shufflevector(e, f, 0, 1, 2, 3, 4, 5, 6, 7);
        } else {
            asm volatile("s_wait_asynccnt 0x0" ::: "memory");   // step-s DMA done
            v4f a = *(const v4f*)(stg);       v4f b = *(const v4f*)(stg + 4);
            v4f e = *(const v4f*)(stg + 256); v4f f = *(const v4f*)(stg + 260);
            c0 = __builtin_shufflevector(a, b, 0, 1, 2, 3, 4, 5, 6, 7);
            c1 = __builtin_shufflevector(e, f, 0, 1, 2, 3, 4, 5, 6, 7);
        }

        // ---- kick next step's XP DMA into the (now consumed) stage buffer ----
        if (s + 1 < S_) {
            asm volatile("s_wait_dscnt 0x0" ::: "memory");      // stage reads done
            const size_t gb = (size_t)(xp_p + (((size_t)(s + 1) * 8 + bt) * 32 + nt0) * 256);
            const unsigned lb = (unsigned)(size_t)(xstage + nt0 * 256);
            #pragma unroll
            for (int j = 0; j < 4; ++j) {
                unsigned long long ga = gb + (unsigned)(lane * 16 + j * 512);
                unsigned           la = lb + (unsigned)(lane * 16 + j * 512);
                asm volatile("global_load_async_to_lds_b128 %0, %1, off"
                             :: "v"(la), "v"(ga) : "memory");
            }
        }

        // ---- c += h @ Wh : shared A per chunk, packed-B b128 per tile ----
        #pragma unroll 4
        for (int k8 = 0; k8 < H_ / 8; ++k8) {
            const int k = k8 * 8;
            v2f a0 = *(const v2f*)(hbuf + hidx(l15, k + 2 * halfsel));
            v2f a1 = *(const v2f*)(hbuf + hidx(l15, k + 4 + 2 * halfsel));
            v4f w0 = *(const v4f*)(wb0 + (size_t)k8 * 4096);
            v4f w1 = *(const v4f*)(wb1 + (size_t)k8 * 4096);
            c0 = __builtin_amdgcn_wmma_f32_16x16x4_f32(false, a0, false,
                    __builtin_shufflevector(w0, w0, 0, 1), (short)0, c0, false, false);
            c0 = __builtin_amdgcn_wmma_f32_16x16x4_f32(false, a1, false,
                    __builtin_shufflevector(w0, w0, 2, 3), (short)0, c0, false, false);
            c1 = __builtin_amdgcn_wmma_f32_16x16x4_f32(false, a0, false,
                    __builtin_shufflevector(w1, w1, 0, 1), (short)0, c1, false, false);
            c1 = __builtin_amdgcn_wmma_f32_16x16x4_f32(false, a1, false,
                    __builtin_shufflevector(w1, w1, 2, 3), (short)0, c1, false, false);
        }

        // ---- h_new = tanh(c) ----
        #pragma unroll
        for (int r = 0; r < 8; ++r) { c0[r] = tanhf(c0[r]); c1[r] = tanhf(c1[r]); }

        __syncthreads();                       // everyone done reading h_{t-1}
        #pragma unroll
        for (int r = 0; r < 8; ++r) {
            const int rm = r + 8 * halfsel;
            hbuf[hidx(rm, n0 + l15)] = c0[r];
            hbuf[hidx(rm, n1 + l15)] = c1[r];
        }
        __syncthreads();                       // h_t visible to all waves
    }

    #pragma unroll
    for (int r = 0; r < 8; ++r) {
        const int rm = r + 8 * halfsel;
        out[(size_t)(bbase + rm) * H_ + n0 + l15] = hbuf[hidx(rm, n0 + l15)];
        out[(size_t)(bbase + rm) * H_ + n1 + l15] = hbuf[hidx(rm, n1 + l15)];
    }
}

extern "C" void kernel_launch(void* const* d_in, const int* in_sizes, int n_in,
                              void* d_out, int out_size, void* d_ws, size_t ws_size,
                              hipStream_t stream) {
    const float* x    = (const float*)d_in[0];  // (128,1024,512)
    const float* Wx   = (const float*)d_in[1];  // (512,512)
    const float* Wh   = (const float*)d_in[2];  // (512,512)
    const float* bias = (const float*)d_in[3];  // (512,)
    float*       out  = (float*)d_out;          // (128,512)

    // Workspace layout (floats): [xp packed: 1024*128*512][Wx packed: 512*512][Wh packed: 512*512]
    float* xp_p = (float*)d_ws;
    float* wxp  = xp_p + (size_t)S_ * B_ * H_;
    float* whp  = wxp + (size_t)D_ * H_;

    pack_w_kernel<<<512, 128, 0, stream>>>(Wx, wxp);
    pack_w_kernel<<<512, 128, 0, stream>>>(Wh, whp);

    dim3 g1(S_, B_ / 16, H_ / 64);              // (1024, 8, 8)
    xp_gemm_kernel<<<g1, 128, 0, stream>>>(x, wxp, bias, xp_p);

    rnn_scan_kernel<<<B_ / 16, 512, 0, stream>>>(xp_p, whp, out);
}